// HGNN_83545703842094
// MI455X (gfx1250) — compile-verified
//
#include <hip/hip_runtime.h>
#include <hip/hip_bf16.h>

// ---------------------------------------------------------------- constants
#define cB   8
#define cN   512
#define cT   8
#define cE   4096
#define cL   64
#define cD   512
#define cH   512
#define cU   256
#define cSQ  22.62741699796952f   // sqrt(512)

typedef __attribute__((ext_vector_type(2))) float v2f;
typedef __attribute__((ext_vector_type(8))) float v8f;

__device__ __forceinline__ float sigmf(float x) { return 1.0f / (1.0f + __expf(-x)); }

// ---------------------------------------------------------------- WMMA GEMM
// C[M,N] = act(alpha * A[M,K] @ op(B) + bias + bias2), batched via blockIdx.z.
// TRANSB=true : B is (N,K) row-major (weight layout, C = A @ W^T)
// TRANSB=false: B is (K,N) row-major
// Each wave computes one 16x16 tile with V_WMMA_F32_16X16X4_F32.
// f32 A frag (16x4): lane L -> row L%16, k-pair (L<16 ? k0,k0+1 : k0+2,k0+3) => contiguous float2
// f32 B frag (4x16): lane L -> col L%16, same k-pair split.
// C/D frag: VGPR i -> row i + (L<16?0:8), col L%16.
template <int ACT, bool TRANSB>
__global__ __launch_bounds__(256) void gemm_wmma_kernel(
    const float* __restrict__ A, const float* __restrict__ Bm,
    const float* __restrict__ bias, const float* __restrict__ bias2,
    float* __restrict__ C, int M, int N, int K, int lda, int ldb, int ldc,
    long long strideA, long long strideB, long long strideC, float alpha) {
  const int lane = threadIdx.x & 31;
  const int wave = threadIdx.x >> 5;
  const int m0 = blockIdx.x * 64 + (wave >> 1) * 16;
  const int n0 = blockIdx.y * 32 + (wave & 1) * 16;
  if (m0 >= M || n0 >= N) return;  // wave-uniform: EXEC stays all-ones for live waves
  const float* Ab = A + (size_t)blockIdx.z * strideA;
  const float* Bb = Bm + (size_t)blockIdx.z * strideB;
  float* Cb = C + (size_t)blockIdx.z * strideC;

  const int l15 = lane & 15;
  const int khalf = (lane < 16) ? 0 : 2;
  const int row = (m0 + l15 < M) ? (m0 + l15) : (M - 1);
  const int col = (n0 + l15 < N) ? (n0 + l15) : (N - 1);

  v8f acc = {};
  for (int k0 = 0; k0 < K; k0 += 4) {
    v2f a = *(const v2f*)(Ab + (size_t)row * lda + k0 + khalf);
    v2f b;
    if (TRANSB) {
      b = *(const v2f*)(Bb + (size_t)col * ldb + k0 + khalf);
    } else {
      b.x = Bb[(size_t)(k0 + khalf) * ldb + col];
      b.y = Bb[(size_t)(k0 + khalf + 1) * ldb + col];
    }
    acc = __builtin_amdgcn_wmma_f32_16x16x4_f32(false, a, false, b, (short)0,
                                                acc, false, false);
  }
  const int rbase = m0 + ((lane < 16) ? 0 : 8);
  const int ccol = n0 + l15;
#pragma unroll
  for (int i = 0; i < 8; ++i) {
    int r = rbase + i;
    if (r < M && ccol < N) {
      float v = acc[i] * alpha;
      if (bias) v += bias[ccol];
      if (bias2) v += bias2[ccol];
      if (ACT == 1) v = fmaxf(v, 0.0f);
      if (ACT == 2) v = sigmf(v);
      Cb[(size_t)r * ldc + ccol] = v;
    }
  }
}

static void gemm(hipStream_t st, const float* A, const float* Bm,
                 const float* bias, const float* bias2, float* C, int M, int N,
                 int K, int lda, int ldb, int ldc, int batch, long long sA,
                 long long sB, long long sC, float alpha, int act, bool transb) {
  dim3 g((M + 63) / 64, (N + 31) / 32, batch), blk(256);
  if (transb) {
    if (act == 0)      gemm_wmma_kernel<0, true><<<g, blk, 0, st>>>(A, Bm, bias, bias2, C, M, N, K, lda, ldb, ldc, sA, sB, sC, alpha);
    else if (act == 1) gemm_wmma_kernel<1, true><<<g, blk, 0, st>>>(A, Bm, bias, bias2, C, M, N, K, lda, ldb, ldc, sA, sB, sC, alpha);
    else               gemm_wmma_kernel<2, true><<<g, blk, 0, st>>>(A, Bm, bias, bias2, C, M, N, K, lda, ldb, ldc, sA, sB, sC, alpha);
  } else {
    if (act == 0)      gemm_wmma_kernel<0, false><<<g, blk, 0, st>>>(A, Bm, bias, bias2, C, M, N, K, lda, ldb, ldc, sA, sB, sC, alpha);
    else if (act == 1) gemm_wmma_kernel<1, false><<<g, blk, 0, st>>>(A, Bm, bias, bias2, C, M, N, K, lda, ldb, ldc, sA, sB, sC, alpha);
    else               gemm_wmma_kernel<2, false><<<g, blk, 0, st>>>(A, Bm, bias, bias2, C, M, N, K, lda, ldb, ldc, sA, sB, sC, alpha);
  }
}

// ---------------------------------------------------------------- misc kernels
__global__ __launch_bounds__(128) void embed_max_kernel(
    const float* __restrict__ emb, const int* __restrict__ idx,
    float* __restrict__ out) {
  const int rowi = blockIdx.x;  // b*N + n
  const int t4 = threadIdx.x;   // 0..127, 4 floats each
  const int base = rowi * cT;
  float4 m = make_float4(-3.4e38f, -3.4e38f, -3.4e38f, -3.4e38f);
  for (int t = 0; t < cT; ++t) {
    const float4 v = *(const float4*)(emb + (size_t)idx[base + t] * cD + t4 * 4);
    m.x = fmaxf(m.x, v.x); m.y = fmaxf(m.y, v.y);
    m.z = fmaxf(m.z, v.z); m.w = fmaxf(m.w, v.w);
  }
  *(float4*)(out + (size_t)rowi * cD + t4 * 4) = m;
}

__global__ __launch_bounds__(128) void embed_copy_kernel(
    const float* __restrict__ emb, const int* __restrict__ idx,
    float* __restrict__ out) {
  const int rowi = blockIdx.x;  // b*L + l
  const int t4 = threadIdx.x;
  const float4 v = *(const float4*)(emb + (size_t)idx[rowi] * cD + t4 * 4);
  *(float4*)(out + (size_t)rowi * cD + t4 * 4) = v;
}

__global__ void zero_kernel(float* __restrict__ p, int n) {
  for (int i = blockIdx.x * blockDim.x + threadIdx.x; i < n; i += gridDim.x * blockDim.x)
    p[i] = 0.0f;
}

__global__ void comp_combine_kernel(const float* __restrict__ Hc,
                                    const float* __restrict__ H1c,
                                    const float* __restrict__ sims,
                                    float* __restrict__ comp, int n) {
  for (int i = blockIdx.x * blockDim.x + threadIdx.x; i < n; i += gridDim.x * blockDim.x) {
    int b = i / (cN * cH);
    comp[i] = Hc[i] + H1c[i] * sims[b];
  }
}

__global__ __launch_bounds__(256) void softmax_rows_kernel(
    float* __restrict__ S, const int* __restrict__ nvalid) {
  const int row = blockIdx.x;
  const int b = row / cN;
  const int nv = nvalid ? nvalid[b] : cN;
  float* sr = S + (size_t)row * cN;
  __shared__ float buf[256];
  const int tid = threadIdx.x;
  float mx = -3.4e38f;
  for (int c = tid; c < cN; c += 256) {
    float v = (c < nv) ? sr[c] : -3.4e38f;
    mx = fmaxf(mx, v);
  }
  buf[tid] = mx; __syncthreads();
  for (int s = 128; s > 0; s >>= 1) { if (tid < s) buf[tid] = fmaxf(buf[tid], buf[tid + s]); __syncthreads(); }
  mx = buf[0]; __syncthreads();
  float sum = 0.0f;
  for (int c = tid; c < cN; c += 256) {
    float v = (c < nv) ? __expf(sr[c] - mx) : 0.0f;
    sr[c] = v; sum += v;
  }
  buf[tid] = sum; __syncthreads();
  for (int s = 128; s > 0; s >>= 1) { if (tid < s) buf[tid] += buf[tid + s]; __syncthreads(); }
  const float inv = 1.0f / buf[0];
  __syncthreads();
  for (int c = tid; c < cN; c += 256) sr[c] *= inv;
}

// biLSTM recurrence, one block per batch row, h/c in LDS, U threads.
__global__ __launch_bounds__(256) void lstm_scan_kernel(
    const float* __restrict__ xg, const float* __restrict__ Whh,
    const float* __restrict__ sims, float* __restrict__ hrev, int reverse) {
  const int b = blockIdx.x;
  const int j = threadIdx.x;  // 0..U-1
  __shared__ float h[cU], c[cU];
  h[j] = 0.0f; c[j] = 0.0f;
  __syncthreads();
  const float sc = sims[b];
  for (int t = 0; t < cL; ++t) {
    const int tt = reverse ? (cL - 1 - t) : t;
    const float* g = xg + ((size_t)b * cL + tt) * (4 * cU);
    float si = g[j], sf = g[j + cU], sg = g[j + 2 * cU], so = g[j + 3 * cU];
    const float* wi = Whh + (size_t)j * cU;
    const float* wf = Whh + (size_t)(j + cU) * cU;
    const float* wg = Whh + (size_t)(j + 2 * cU) * cU;
    const float* wo = Whh + (size_t)(j + 3 * cU) * cU;
    for (int k = 0; k < cU; ++k) {
      const float hk = h[k];
      si += hk * wi[k]; sf += hk * wf[k]; sg += hk * wg[k]; so += hk * wo[k];
    }
    const float cn = sigmf(sf) * c[j] + sigmf(si) * tanhf(sg);
    const float hn = sigmf(so) * tanhf(cn);
    __syncthreads();
    h[j] = hn; c[j] = cn;
    __syncthreads();
    hrev[((size_t)b * cL + tt) * (2 * cU) + j + (reverse ? cU : 0)] = hn * sc;
  }
}

// contrib[b,e] = (1/sqrt(H)) * sum_d q[b,src,d] * relu(attr . WR[d,:]) -> atomic add into A
__global__ __launch_bounds__(128) void contrib_scatter_kernel(
    const float* __restrict__ q2, const float* __restrict__ attr,
    const float* __restrict__ WR, const int* __restrict__ eidx,
    float* __restrict__ S) {
  const int e = blockIdx.x;  // b*E + le
  const int b = e / cE, le = e % cE;
  const int* ei = eidx + (size_t)b * 2 * cE;
  const int src = ei[le], dst = ei[cE + le];
  const float* a3 = attr + ((size_t)b * cE + le) * 3;
  const float a0 = a3[0], a1 = a3[1], a2 = a3[2];
  const float* qrow = q2 + ((size_t)b * cN + src) * cH;
  float acc = 0.0f;
  for (int d = threadIdx.x; d < cH; d += 128) {
    const float ead = fmaxf(a0 * WR[d * 3] + a1 * WR[d * 3 + 1] + a2 * WR[d * 3 + 2], 0.0f);
    acc += qrow[d] * ead;
  }
  __shared__ float buf[128];
  buf[threadIdx.x] = acc; __syncthreads();
  for (int s = 64; s > 0; s >>= 1) { if (threadIdx.x < s) buf[threadIdx.x] += buf[threadIdx.x + s]; __syncthreads(); }
  if (threadIdx.x == 0)
    atomicAdd(&S[((size_t)b * cN + src) * cN + dst], buf[0] * (1.0f / cSQ));
}

__global__ void norm_gather_kernel(const float* __restrict__ S,
                                   const int* __restrict__ eidx,
                                   float* __restrict__ normv, int n) {
  for (int e = blockIdx.x * blockDim.x + threadIdx.x; e < n; e += gridDim.x * blockDim.x) {
    const int b = e / cE, le = e % cE;
    const int* ei = eidx + (size_t)b * 2 * cE;
    const int src = ei[le], dst = ei[cE + le];
    normv[e] = S[((size_t)b * cN + src) * cN + dst];
  }
}

__global__ __launch_bounds__(128) void seg_hv_kernel(
    const float* __restrict__ x0, const int* __restrict__ eidx,
    float* __restrict__ hv) {
  const int e = blockIdx.x;
  const int b = e / cE, le = e % cE;
  const int* ei = eidx + (size_t)b * 2 * cE;
  const int src = ei[le], dst = ei[cE + le];
  const float* xs = x0 + ((size_t)b * cN + src) * cH;
  float* hd = hv + ((size_t)b * cN + dst) * cH;
  for (int d = threadIdx.x; d < cH; d += 128) atomicAdd(&hd[d], xs[d]);
}

__global__ __launch_bounds__(128) void msg_seg_kernel(
    const float* __restrict__ WVx, const float* __restrict__ attr,
    const float* __restrict__ WF, const float* __restrict__ normv,
    const int* __restrict__ eidx, float* __restrict__ hv_) {
  const int e = blockIdx.x;
  const int b = e / cE, le = e % cE;
  const int* ei = eidx + (size_t)b * 2 * cE;
  const int src = ei[le], dst = ei[cE + le];
  const float* a3 = attr + ((size_t)b * cE + le) * 3;
  const float a0 = a3[0], a1 = a3[1], a2 = a3[2];
  const float nv = normv[e];
  const float* vs = WVx + ((size_t)b * cN + src) * cH;
  float* hd = hv_ + ((size_t)b * cN + dst) * cH;
  for (int d = threadIdx.x; d < cH; d += 128) {
    const float eaf = a0 * WF[d * 3] + a1 * WF[d * 3 + 1] + a2 * WF[d * 3 + 2];
    atomicAdd(&hd[d], nv * (vs[d] + eaf));
  }
}

__global__ void build_cat_kernel(const float* __restrict__ a,
                                 const float* __restrict__ b,
                                 float* __restrict__ cat, int n) {
  for (int i = blockIdx.x * blockDim.x + threadIdx.x; i < n; i += gridDim.x * blockDim.x) {
    const int m = i / cH, j = i % cH;
    const float av = a[i], bv = b[i];
    float* cr = cat + (size_t)m * (4 * cH) + j;
    cr[0] = av; cr[cH] = bv; cr[2 * cH] = av * bv; cr[3 * cH] = av - bv;
  }
}

__global__ void fuse_apply_kernel(const float* __restrict__ z,
                                  const float* __restrict__ a,
                                  const float* __restrict__ b,
                                  float* __restrict__ out, int n) {
  for (int i = blockIdx.x * blockDim.x + threadIdx.x; i < n; i += gridDim.x * blockDim.x) {
    const float zz = z[i];
    out[i] = zz * a[i] + (1.0f - zz) * b[i];
  }
}

__global__ void gru_apply_kernel(const float* __restrict__ gi,
                                 const float* __restrict__ gh,
                                 const float* __restrict__ fprev,
                                 float* __restrict__ fv, int n) {
  for (int i = blockIdx.x * blockDim.x + threadIdx.x; i < n; i += gridDim.x * blockDim.x) {
    const int m = i / cH, j = i % cH;
    const float* gir = gi + (size_t)m * (3 * cH);
    const float* ghr = gh + (size_t)m * (3 * cH);
    const float r = sigmf(gir[j] + ghr[j]);
    const float zz = sigmf(gir[cH + j] + ghr[cH + j]);
    const float nn = tanhf(gir[2 * cH + j] + r * ghr[2 * cH + j]);
    fv[i] = (1.0f - zz) * nn + zz * fprev[i];
  }
}

__global__ __launch_bounds__(128) void rowmax_kernel(const float* __restrict__ fv,
                                                     float* __restrict__ greps) {
  const int b = blockIdx.x / (cH / 128);
  const int j = (blockIdx.x % (cH / 128)) * 128 + threadIdx.x;
  float m = -3.4e38f;
  for (int n = 0; n < cN; ++n) m = fmaxf(m, fv[((size_t)b * cN + n) * cH + j]);
  greps[(size_t)b * cH + j] = m;
}

__global__ void revfinal_kernel(const float* __restrict__ hrev,
                                float* __restrict__ revf, int n) {
  for (int i = blockIdx.x * blockDim.x + threadIdx.x; i < n; i += gridDim.x * blockDim.x) {
    const int b = i / cH, j = i % cH;
    revf[i] = hrev[((size_t)b * cL + (cL - 1)) * cH + j];
  }
}

__global__ void final_small_kernel(const float* __restrict__ z2,
                                   const float* __restrict__ greps,
                                   const float* __restrict__ revf,
                                   float* __restrict__ out, int n) {
  const size_t enc = (size_t)cB * (cN + cL) * cH;
  for (int i = blockIdx.x * blockDim.x + threadIdx.x; i < n; i += gridDim.x * blockDim.x) {
    const float zz = z2[i];
    out[enc + i] = zz * greps[i] + (1.0f - zz) * revf[i];  // init_h
    out[enc + (size_t)cB * cH + i] = 0.0f;                 // init_c (zeros)
  }
}

__global__ void enc_out_kernel(const float* __restrict__ fv,
                               const float* __restrict__ hrev,
                               float* __restrict__ out, int n) {
  for (int i = blockIdx.x * blockDim.x + threadIdx.x; i < n; i += gridDim.x * blockDim.x) {
    const int per = (cN + cL) * cH;
    const int b = i / per, rem = i % per;
    const int nn = rem / cH, j = rem % cH;
    out[i] = (nn < cN) ? fv[((size_t)b * cN + nn) * cH + j]
                       : hrev[((size_t)b * cL + (nn - cN)) * cH + j];
  }
}

// ---------------------------------------------------------------- driver
extern "C" void kernel_launch(void* const* d_in, const int* in_sizes, int n_in,
                              void* d_out, int out_size, void* d_ws,
                              size_t ws_size, hipStream_t stream) {
  (void)in_sizes; (void)n_in; (void)out_size; (void)ws_size;
  const int* batch_nodes = (const int*)d_in[0];
  const int* rev_batch   = (const int*)d_in[1];
  const int* rev_nnum    = (const int*)d_in[2];
  const int* rev_coms    = (const int*)d_in[3];
  const float* sims      = (const float*)d_in[5];
  const int* edge_index  = (const int*)d_in[6];
  const float* edge_attr = (const float*)d_in[7];
  const float* code_emb  = (const float*)d_in[8];
  const float* nl_emb    = (const float*)d_in[9];
  const float* Wih_f = (const float*)d_in[10];
  const float* Whh_f = (const float*)d_in[11];
  const float* bih_f = (const float*)d_in[12];
  const float* bhh_f = (const float*)d_in[13];
  const float* Wih_b = (const float*)d_in[14];
  const float* Whh_b = (const float*)d_in[15];
  const float* bih_b = (const float*)d_in[16];
  const float* bhh_b = (const float*)d_in[17];
  const float* Wq = (const float*)d_in[18];
  const float* Wk = (const float*)d_in[19];
  const float* pWQ = (const float*)d_in[20];
  const float* pWK = (const float*)d_in[21];
  const float* pWR = (const float*)d_in[22];
  const float* pWV = (const float*)d_in[23];
  const float* pWF = (const float*)d_in[24];
  const float* gWih = (const float*)d_in[25];
  const float* gWhh = (const float*)d_in[26];
  const float* gbih = (const float*)d_in[27];
  const float* gbhh = (const float*)d_in[28];
  const float* Wz = (const float*)d_in[29];
  const float* bz = (const float*)d_in[30];
  float* out = (float*)d_out;

  float* ws = (float*)d_ws;
  size_t o = 0;
  auto alloc = [&](size_t nf) { float* p = ws + o; o += nf; return p; };
  const size_t MH = (size_t)cB * cN * cH;           // 2,097,152
  float* Hc   = alloc(MH);                          // later: WVx
  float* Hc2  = alloc(MH);                          // later: hv_
  float* qA   = alloc(MH);                          // later: q2
  float* kA   = alloc(MH);                          // later: k2
  float* S    = alloc(MH);                          // (B,N,N)
  float* H1c  = alloc(MH);                          // later: hv
  float* comp = alloc(MH);
  float* fv   = alloc(MH);
  float* xnl  = alloc((size_t)cB * cL * cD);
  float* xgf  = alloc((size_t)cB * cL * 4 * cU);
  float* xgb  = alloc((size_t)cB * cL * 4 * cU);
  float* hrev = alloc((size_t)cB * cL * cH);
  float* normv= alloc((size_t)cB * cE);
  float* catb = alloc(MH * 4);                      // (B*N, 4H), reused for cat2
  float* zb   = alloc(MH);                          // z / z2
  float* fuseb= alloc(MH);
  float* gi   = alloc(MH * 3);
  float* gh   = alloc(MH * 3);
  float* greps= alloc((size_t)cB * cH);
  float* revf = alloc((size_t)cB * cH);

  const int M = cB * cN;              // 4096
  const long long sAtt = (long long)cN * cN;  // 262144 (per-batch N*N and N*H both = 262144)

  // 1) embedding gather+max
  embed_max_kernel<<<cB * cN, 128, 0, stream>>>(code_emb, batch_nodes, Hc);
  embed_max_kernel<<<cB * cN, 128, 0, stream>>>(code_emb, rev_batch, Hc2);
  // 2-3) q = Hc@Wq^T, k = Hc_@Wk^T   (fp32 WMMA GEMMs)
  gemm(stream, Hc, Wq, nullptr, nullptr, qA, M, cH, cH, cH, cH, cH, 1, 0, 0, 0, 1.0f, 0, true);
  gemm(stream, Hc2, Wk, nullptr, nullptr, kA, M, cH, cH, cH, cH, cH, 1, 0, 0, 0, 1.0f, 0, true);
  // 4) score = q @ k^T (batched)
  gemm(stream, qA, kA, nullptr, nullptr, S, cN, cN, cH, cH, cH, cN, cB, sAtt, sAtt, sAtt, 1.0f, 0, true);
  // 5) masked softmax
  softmax_rows_kernel<<<cB * cN, 256, 0, stream>>>(S, rev_nnum);
  // 6) H1c = P @ Hc_ (batched, non-transposed B)
  gemm(stream, S, Hc2, nullptr, nullptr, H1c, cN, cH, cN, cN, cH, cH, cB, sAtt, sAtt, sAtt, 1.0f, 0, false);
  // 7) comp = Hc + H1c * sims
  comp_combine_kernel<<<2048, 256, 0, stream>>>(Hc, H1c, sims, comp, (int)MH);
  // 8) biLSTM: input gates via WMMA GEMM, recurrence in LDS
  embed_copy_kernel<<<cB * cL, 128, 0, stream>>>(nl_emb, rev_coms, xnl);
  gemm(stream, xnl, Wih_f, bih_f, bhh_f, xgf, cB * cL, 4 * cU, cD, cD, cD, 4 * cU, 1, 0, 0, 0, 1.0f, 0, true);
  gemm(stream, xnl, Wih_b, bih_b, bhh_b, xgb, cB * cL, 4 * cU, cD, cD, cD, 4 * cU, 1, 0, 0, 0, 1.0f, 0, true);
  lstm_scan_kernel<<<cB, cU, 0, stream>>>(xgf, Whh_f, sims, hrev, 0);
  lstm_scan_kernel<<<cB, cU, 0, stream>>>(xgb, Whh_b, sims, hrev, 1);
  // 10) q2 = relu(comp@WQ^T), k2 = relu(comp@WK^T)  (reuse qA/kA)
  gemm(stream, comp, pWQ, nullptr, nullptr, qA, M, cH, cH, cH, cH, cH, 1, 0, 0, 0, 1.0f, 1, true);
  gemm(stream, comp, pWK, nullptr, nullptr, kA, M, cH, cH, cH, cH, cH, 1, 0, 0, 0, 1.0f, 1, true);
  // 11) base = q2 @ k2^T / sqrt(H)
  gemm(stream, qA, kA, nullptr, nullptr, S, cN, cN, cH, cH, cH, cN, cB, sAtt, sAtt, sAtt, 1.0f / cSQ, 0, true);
  // 12) scatter edge contributions, 13) softmax, 14) gather norms
  contrib_scatter_kernel<<<cB * cE, 128, 0, stream>>>(qA, edge_attr, pWR, edge_index, S);
  softmax_rows_kernel<<<cB * cN, 256, 0, stream>>>(S, nullptr);
  norm_gather_kernel<<<128, 256, 0, stream>>>(S, edge_index, normv, cB * cE);
  // 15) WVx = comp @ WV^T (into Hc slot)
  gemm(stream, comp, pWV, nullptr, nullptr, Hc, M, cH, cH, cH, cH, cH, 1, 0, 0, 0, 1.0f, 0, true);
  // 16-18) segment sums (hv into H1c slot, hv_ into Hc2 slot)
  zero_kernel<<<2048, 256, 0, stream>>>(H1c, (int)MH);
  zero_kernel<<<2048, 256, 0, stream>>>(Hc2, (int)MH);
  seg_hv_kernel<<<cB * cE, 128, 0, stream>>>(comp, edge_index, H1c);
  msg_seg_kernel<<<cB * cE, 128, 0, stream>>>(Hc, edge_attr, pWF, normv, edge_index, Hc2);
  // 19-21) gated fuse: z = sigmoid(cat@Wz^T + bz); fuse = z*hv + (1-z)*hv_
  build_cat_kernel<<<2048, 256, 0, stream>>>(H1c, Hc2, catb, (int)MH);
  gemm(stream, catb, Wz, bz, nullptr, zb, M, cH, 4 * cH, 4 * cH, 4 * cH, cH, 1, 0, 0, 0, 1.0f, 2, true);
  fuse_apply_kernel<<<2048, 256, 0, stream>>>(zb, H1c, Hc2, fuseb, (int)MH);
  // 22-24) GRU cell
  gemm(stream, fuseb, gWih, gbih, nullptr, gi, M, 3 * cH, cH, cH, cH, 3 * cH, 1, 0, 0, 0, 1.0f, 0, true);
  gemm(stream, comp, gWhh, gbhh, nullptr, gh, M, 3 * cH, cH, cH, cH, 3 * cH, 1, 0, 0, 0, 1.0f, 0, true);
  gru_apply_kernel<<<2048, 256, 0, stream>>>(gi, gh, comp, fv, (int)MH);
  // 25-27) graph readout + init_h fuse (M=8 tile, bounds-checked in GEMM)
  rowmax_kernel<<<cB * (cH / 128), 128, 0, stream>>>(fv, greps);
  revfinal_kernel<<<16, 256, 0, stream>>>(hrev, revf, cB * cH);
  build_cat_kernel<<<16, 256, 0, stream>>>(greps, revf, catb, cB * cH);
  gemm(stream, catb, Wz, bz, nullptr, zb, cB, cH, 4 * cH, 4 * cH, 4 * cH, cH, 1, 0, 0, 0, 1.0f, 2, true);
  // 28-29) outputs
  final_small_kernel<<<16, 256, 0, stream>>>(zb, greps, revf, out, cB * cH);
  enc_out_kernel<<<2304, 256, 0, stream>>>(fv, hrev, out, cB * (cN + cL) * cH);
}